// SparseTemporalAttention_36447092474235
// MI455X (gfx1250) — compile-verified
//
#include <hip/hip_runtime.h>
#include <hip/hip_bf16.h>

typedef __attribute__((ext_vector_type(2))) float v2f;
typedef __attribute__((ext_vector_type(8))) float v8f;

#define B_ 4
#define T_ 4096
#define D_ 512
#define KKEEP 1024
#define INV_SQRT_D 0.04419417382415922f

// ---------------------------------------------------------------------------
// K1: q_b = h[b,T-1,:] @ Wq + bq  ->  Qpad[16][512] (rows 4..15 zero)
//     cb[b] = bk . q_b
// grid(16), block(512)
// ---------------------------------------------------------------------------
__global__ void sta_qproj(const float* __restrict__ h, const float* __restrict__ Wq,
                          const float* __restrict__ bq, const float* __restrict__ bk,
                          float* __restrict__ Qpad, float* __restrict__ cb) {
    __shared__ float red[512];
    const int b = blockIdx.x;
    const int d = threadIdx.x;
    if (b < B_) {
        const float* hrow = h + ((size_t)b * T_ + (T_ - 1)) * D_;
        float acc = bq[d];
        #pragma unroll 4
        for (int j = 0; j < D_; ++j)
            acc += hrow[j] * Wq[j * D_ + d];          // coalesced over d
        Qpad[b * D_ + d] = acc;
        red[d] = bk[d] * acc;
        __syncthreads();
        for (int s = 256; s > 0; s >>= 1) {
            if (d < s) red[d] += red[d + s];
            __syncthreads();
        }
        if (d == 0) cb[b] = red[0];
    } else {
        Qpad[b * D_ + d] = 0.0f;
    }
}

// ---------------------------------------------------------------------------
// K2: Wt[n][d] = sum_j Wk[d][j] * Qpad[n][j]   ( = (Wk q_n)[d] )
// fp32 WMMA 16x16x4: A = Wk row-tile (16xK), B = Qpad^T (Kx16). One wave/block.
// grid(32), block(32)
// ---------------------------------------------------------------------------
__global__ void sta_wk_gemm(const float* __restrict__ Wk, const float* __restrict__ Qpad,
                            float* __restrict__ Wt) {
    const int lane = threadIdx.x;
    const int d0   = blockIdx.x * 16;
    const int m    = lane & 15;            // A row / B col within tile
    const int k0   = (lane >> 4) << 1;     // K pair owned by this lane half
    v8f c = {};
    for (int j = 0; j < D_; j += 4) {
        v2f a, bb;
        a.x  = Wk[(d0 + m) * D_ + j + k0];
        a.y  = Wk[(d0 + m) * D_ + j + k0 + 1];
        bb.x = Qpad[m * D_ + j + k0];      // B[k][n] = Qpad[n][k]
        bb.y = Qpad[m * D_ + j + k0 + 1];
        c = __builtin_amdgcn_wmma_f32_16x16x4_f32(false, a, false, bb,
                                                  (short)0, c, false, false);
    }
    // D tile: lanes 0-15 -> rows v (N=lane); lanes 16-31 -> rows 8+v (N=lane-16)
    const int mbase = (lane >> 4) * 8;
    #pragma unroll
    for (int v = 0; v < 8; ++v)
        Wt[m * D_ + d0 + mbase + v] = c[v];   // store transposed: Wt[n][d]
}

// ---------------------------------------------------------------------------
// K3: s[b*T + t] = (h[b,t,:] . Wt[b,:] + cb[b]) / sqrt(D)
// One wave per row; w_b cached in LDS; fully coalesced stream of h (32 MB).
// grid(B_*T_/8), block(256)
// ---------------------------------------------------------------------------
__global__ void sta_scores(const float* __restrict__ h, const float* __restrict__ Wt,
                           const float* __restrict__ cb, float* __restrict__ s) {
    __shared__ float w[D_];
    const int tid  = threadIdx.x;
    const int row0 = blockIdx.x * 8;
    const int b    = row0 >> 12;           // rows per batch = 4096
    for (int i = tid; i < D_; i += 256) w[i] = Wt[b * D_ + i];
    __syncthreads();
    const int wave = tid >> 5, lane = tid & 31;
    const int row  = row0 + wave;
    const float* hr = h + (size_t)row * D_;
    float acc = 0.0f;
    #pragma unroll
    for (int c = 0; c < 4; ++c) {
        const int idx = c * 128 + lane * 4;
        float4 hv = *(const float4*)(hr + idx);
        float4 wv = *(const float4*)(&w[idx]);
        acc += hv.x * wv.x + hv.y * wv.y + hv.z * wv.z + hv.w * wv.w;
    }
    #pragma unroll
    for (int off = 16; off > 0; off >>= 1) acc += __shfl_xor(acc, off, 32);
    if (lane == 0) s[row] = (acc + cb[b]) * INV_SQRT_D;
}

// ---------------------------------------------------------------------------
// K4: per-batch top-k threshold via bitonic sort (descending) in LDS, then
// dense softmax-numerator weights aw[t] (0 if not selected) + denom (tree sum).
// grid(4), block(1024)
// ---------------------------------------------------------------------------
__global__ void sta_topk(const float* __restrict__ s, float* __restrict__ aw,
                         float* __restrict__ denom) {
    __shared__ float sv[T_];
    __shared__ float red[1024];
    const int b = blockIdx.x, tid = threadIdx.x;
    const float* sb = s + b * T_;
    for (int i = tid; i < T_; i += 1024) sv[i] = sb[i];
    __syncthreads();
    // bitonic sort, descending
    for (int k = 2; k <= T_; k <<= 1) {
        for (int j = k >> 1; j > 0; j >>= 1) {
            for (int i = tid; i < T_; i += 1024) {
                const int ixj = i ^ j;
                if (ixj > i) {
                    const float a = sv[i], c = sv[ixj];
                    const bool descBlock = ((i & k) == 0);
                    if (descBlock ? (a < c) : (a > c)) { sv[i] = c; sv[ixj] = a; }
                }
            }
            __syncthreads();
        }
    }
    const float theta = sv[KKEEP - 1];
    const float mx    = sv[0];
    float local = 0.0f;
    for (int i = tid; i < T_; i += 1024) {
        const float val = sb[i];
        const float e = (val >= theta) ? expf(val - mx) : 0.0f;
        aw[b * T_ + i] = e;
        local += e;
    }
    red[tid] = local;
    __syncthreads();
    for (int st = 512; st > 0; st >>= 1) {
        if (tid < st) red[tid] += red[tid + st];
        __syncthreads();
    }
    if (tid == 0) denom[b] = red[0];
}

// ---------------------------------------------------------------------------
// K5: Upad[b][d] = (1/denom[b]) * sum_t aw[t] * h[b,t,d]   (rows 4..15 zero)
// Only selected rows (~25%) are actually loaded from HBM (uniform branch).
// grid(4 colblocks, 16 brows), block(128)
// ---------------------------------------------------------------------------
__global__ void sta_wsum(const float* __restrict__ h, const float* __restrict__ aw,
                         const float* __restrict__ denom, float* __restrict__ Upad) {
    const int cb0 = blockIdx.x * 128;
    const int b   = blockIdx.y;
    const int tid = threadIdx.x;
    if (b >= B_) { Upad[b * D_ + cb0 + tid] = 0.0f; return; }
    __shared__ float w[T_];
    for (int i = tid; i < T_; i += 128) w[i] = aw[b * T_ + i];
    __syncthreads();
    const float* hb = h + (size_t)b * T_ * D_;
    float acc = 0.0f;
    for (int t = 0; t < T_; ++t) {
        const float wt = w[t];                 // uniform across block
        if (wt != 0.0f) acc += wt * hb[(size_t)t * D_ + cb0 + tid];
    }
    Upad[b * D_ + cb0 + tid] = acc / denom[b];
}

// ---------------------------------------------------------------------------
// K6: out[b][n] = Upad[b,:] @ Wv[:,n] + bv[n], via fp32 WMMA 16x16x4.
// grid(32), block(32)
// ---------------------------------------------------------------------------
__global__ void sta_out_gemm(const float* __restrict__ Upad, const float* __restrict__ Wv,
                             const float* __restrict__ bv, float* __restrict__ out) {
    const int lane = threadIdx.x;
    const int n0   = blockIdx.x * 16;
    const int m    = lane & 15;
    const int k0   = (lane >> 4) << 1;
    v8f c = {};
    for (int j = 0; j < D_; j += 4) {
        v2f a, bb;
        a.x  = Upad[m * D_ + j + k0];
        a.y  = Upad[m * D_ + j + k0 + 1];
        bb.x = Wv[(j + k0) * D_ + n0 + m];
        bb.y = Wv[(j + k0 + 1) * D_ + n0 + m];
        c = __builtin_amdgcn_wmma_f32_16x16x4_f32(false, a, false, bb,
                                                  (short)0, c, false, false);
    }
    if (lane < 16) {                       // only rows 0..3 are real batches
        #pragma unroll
        for (int v = 0; v < B_; ++v)
            out[v * D_ + n0 + lane] = c[v] + bv[n0 + lane];
    }
}

extern "C" void kernel_launch(void* const* d_in, const int* in_sizes, int n_in,
                              void* d_out, int out_size, void* d_ws, size_t ws_size,
                              hipStream_t stream) {
    (void)in_sizes; (void)n_in; (void)out_size; (void)ws_size;
    const float* h  = (const float*)d_in[0];
    const float* Wq = (const float*)d_in[1];
    const float* bq = (const float*)d_in[2];
    const float* Wk = (const float*)d_in[3];
    const float* bk = (const float*)d_in[4];
    const float* Wv = (const float*)d_in[5];
    const float* bv = (const float*)d_in[6];

    float* ws    = (float*)d_ws;
    float* Qpad  = ws;              // 16*512
    float* Wt    = ws + 8192;       // 16*512 (transposed: [n][d])
    float* cb    = ws + 16384;      // 16
    float* s     = ws + 16400;      // 4*4096
    float* aw    = ws + 32784;      // 4*4096
    float* denom = ws + 49168;      // 4 (+pad)
    float* Upad  = ws + 49184;      // 16*512

    sta_qproj   <<<16, 512, 0, stream>>>(h, Wq, bq, bk, Qpad, cb);
    sta_wk_gemm <<<32, 32, 0, stream>>>(Wk, Qpad, Wt);
    sta_scores  <<<(B_ * T_) / 8, 256, 0, stream>>>(h, Wt, cb, s);
    sta_topk    <<<B_, 1024, 0, stream>>>(s, aw, denom);
    sta_wsum    <<<dim3(4, 16), 128, 0, stream>>>(h, aw, denom, Upad);
    sta_out_gemm<<<32, 32, 0, stream>>>(Upad, Wv, bv, (float*)d_out);
}